// FlameLandmarks_3650722202071
// MI455X (gfx1250) — compile-verified
//
#include <hip/hip_runtime.h>
#include <hip/hip_bf16.h>

typedef __attribute__((ext_vector_type(2))) float v2f;
typedef __attribute__((ext_vector_type(8))) float v8f;

#define FB   1024
#define FV   5023
#define N3   15069          // V*3
#define FNJ  5
#define FNL  51
#define NMAC 471            // ceil(15069/32)
#define MMAC 32             // 1024/32
#define LM_OFF  15430656    // B*N3
#define REG_OFF 15587328    // LM_OFF + B*NL*3

// ---------------- kernel A: Jt = Jreg@v_template, Js = Jreg@shapedirs ----------------
__global__ void flame_kA(const float* __restrict__ Jreg, const float* __restrict__ vt,
                         const float* __restrict__ sd, float* __restrict__ Js,
                         float* __restrict__ Jt) {
    int i = blockIdx.x * blockDim.x + threadIdx.x;
    if (i < 6000) {                       // Js[(j*3+k)*400 + l]
        int jk = i / 400, l = i % 400;
        int j = jk / 3, k = jk % 3;
        const float* jr = Jreg + (size_t)j * FV;
        float acc = 0.f;
        for (int v = 0; v < FV; ++v)
            acc += jr[v] * sd[((size_t)v * 3 + k) * 400 + l];
        Js[i] = acc;
    } else if (i < 6015) {                // Jt[j*3+k]
        int jk = i - 6000;
        int j = jk / 3, k = jk % 3;
        const float* jr = Jreg + (size_t)j * FV;
        float acc = 0.f;
        for (int v = 0; v < FV; ++v)
            acc += jr[v] * vt[v * 3 + k];
        Jt[jk] = acc;
    }
}

// ---------------- kernel B: per-batch joints, Rodrigues, chain, A, pose_feature ------
__global__ void flame_kB(const float* __restrict__ sp, const float* __restrict__ ep,
                         const float* __restrict__ grot, const float* __restrict__ neck,
                         const float* __restrict__ jaw, const float* __restrict__ eye,
                         const float* __restrict__ Js, const float* __restrict__ Jt,
                         float* __restrict__ pf, float* __restrict__ Aws) {
    int b = blockIdx.x * blockDim.x + threadIdx.x;
    if (b >= FB) return;

    const float* sb = sp + (size_t)b * 300;
    const float* eb = ep + (size_t)b * 100;

    // joints = Jt + Js . betas
    float jnt[FNJ][3];
    for (int j = 0; j < FNJ; ++j)
        for (int k = 0; k < 3; ++k) {
            const float* row = Js + (size_t)(j * 3 + k) * 400;
            float acc = Jt[j * 3 + k];
            for (int l = 0; l < 300; ++l) acc += row[l] * sb[l];
            for (int l = 0; l < 100; ++l) acc += row[300 + l] * eb[l];
            jnt[j][k] = acc;
        }

    // full pose (15)
    float p[15];
    p[0] = grot[b * 3 + 0]; p[1] = grot[b * 3 + 1]; p[2] = grot[b * 3 + 2];
    p[3] = neck[b * 3 + 0]; p[4] = neck[b * 3 + 1]; p[5] = neck[b * 3 + 2];
    p[6] = jaw [b * 3 + 0]; p[7] = jaw [b * 3 + 1]; p[8] = jaw [b * 3 + 2];
    for (int i = 0; i < 6; ++i) p[9 + i] = eye[b * 6 + i];

    // Rodrigues
    float R[FNJ][9];
    for (int j = 0; j < FNJ; ++j) {
        float r0 = p[3 * j], r1 = p[3 * j + 1], r2 = p[3 * j + 2];
        float a0 = r0 + 1e-8f, a1 = r1 + 1e-8f, a2 = r2 + 1e-8f;
        float ang = sqrtf(a0 * a0 + a1 * a1 + a2 * a2);
        float inv = 1.f / ang;
        float ax = r0 * inv, ay = r1 * inv, az = r2 * inv;
        float s = sinf(ang), c = cosf(ang), t1 = 1.f - c;
        float K[9] = { 0.f, -az,  ay,
                       az,  0.f, -ax,
                      -ay,  ax,  0.f };
        float K2[9];
        for (int r = 0; r < 3; ++r)
            for (int cc = 0; cc < 3; ++cc) {
                float acc = 0.f;
                for (int t = 0; t < 3; ++t) acc += K[r * 3 + t] * K[t * 3 + cc];
                K2[r * 3 + cc] = acc;
            }
        for (int idx = 0; idx < 9; ++idx)
            R[j][idx] = ((idx % 4 == 0) ? 1.f : 0.f) + s * K[idx] + t1 * K2[idx];
    }

    // pose feature (rot_mats[1:] - I)
    for (int j = 1; j < FNJ; ++j)
        for (int idx = 0; idx < 9; ++idx)
            pf[(size_t)b * 36 + (j - 1) * 9 + idx] = R[j][idx] - ((idx % 4 == 0) ? 1.f : 0.f);

    // relative joints, kinematic chain
    const int par[FNJ] = { -1, 0, 1, 1, 1 };
    float rel[FNJ][3];
    for (int k = 0; k < 3; ++k) rel[0][k] = jnt[0][k];
    for (int j = 1; j < FNJ; ++j)
        for (int k = 0; k < 3; ++k) rel[j][k] = jnt[j][k] - jnt[par[j]][k];

    float ch[FNJ][16];
    for (int r = 0; r < 3; ++r) {
        for (int cc = 0; cc < 3; ++cc) ch[0][r * 4 + cc] = R[0][r * 3 + cc];
        ch[0][r * 4 + 3] = rel[0][r];
    }
    ch[0][12] = 0.f; ch[0][13] = 0.f; ch[0][14] = 0.f; ch[0][15] = 1.f;

    for (int j = 1; j < FNJ; ++j) {
        float tf[16];
        for (int r = 0; r < 3; ++r) {
            for (int cc = 0; cc < 3; ++cc) tf[r * 4 + cc] = R[j][r * 3 + cc];
            tf[r * 4 + 3] = rel[j][r];
        }
        tf[12] = 0.f; tf[13] = 0.f; tf[14] = 0.f; tf[15] = 1.f;
        const float* cp = ch[par[j]];
        for (int r = 0; r < 4; ++r)
            for (int cc = 0; cc < 4; ++cc) {
                float acc = 0.f;
                for (int t = 0; t < 4; ++t) acc += cp[r * 4 + t] * tf[t * 4 + cc];
                ch[j][r * 4 + cc] = acc;
            }
    }

    // A = chain with translation -= R_chain @ joints ; store rows 0..2 (3x4)
    for (int j = 0; j < FNJ; ++j)
        for (int r = 0; r < 3; ++r) {
            float jh = ch[j][r * 4 + 0] * jnt[j][0] + ch[j][r * 4 + 1] * jnt[j][1] +
                       ch[j][r * 4 + 2] * jnt[j][2];
            float* dst = Aws + (size_t)b * 60 + j * 12 + r * 4;
            dst[0] = ch[j][r * 4 + 0];
            dst[1] = ch[j][r * 4 + 1];
            dst[2] = ch[j][r * 4 + 2];
            dst[3] = ch[j][r * 4 + 3] - jh;
        }
}

// ---------------- kernel C: WMMA GEMM, 32x32 macro-tile (2M x 2N register blocking) --
// v_posed = v_template + betas@shapedirs^T + pose_feature@posedirs
__global__ __launch_bounds__(128) void flame_kC(
        const float* __restrict__ sp, const float* __restrict__ ep,
        const float* __restrict__ sd, const float* __restrict__ pd,
        const float* __restrict__ pf, const float* __restrict__ vt,
        float* __restrict__ out) {
    int wid  = threadIdx.x >> 5;
    int lane = threadIdx.x & 31;
    int tile = blockIdx.x * 4 + wid;          // 32*471 = 15072 macro tiles
    int mt = tile / NMAC, nt = tile % NMAC;
    int m0 = mt * 32, n0 = nt * 32;
    int laneN = lane & 15;
    int hi = lane >> 4;                        // 0: K=k0,k0+1 ; 1: K=k0+2,k0+3
    int na = n0 + laneN;                       // N subtile 0 column
    int nb = na + 16;                          // N subtile 1 column
    bool aOK = na < N3, bOK = nb < N3;
    int nca = aOK ? na : (N3 - 1);
    int ncb = bOK ? nb : (N3 - 1);
    const float* sdrowA = sd + (size_t)nca * 400;
    const float* sdrowB = sd + (size_t)ncb * 400;
    int rowA0 = m0 + laneN;                    // M subtile 0 row (always < 1024)
    int rowA1 = rowA0 + 16;                    // M subtile 1 row
    const float* as0 = sp + (size_t)rowA0 * 300 + 2 * hi;
    const float* as1 = sp + (size_t)rowA1 * 300 + 2 * hi;
    const float* ae0 = ep + (size_t)rowA0 * 100 + 2 * hi;
    const float* ae1 = ep + (size_t)rowA1 * 100 + 2 * hi;
    const float* ap0 = pf + (size_t)rowA0 * 36  + 2 * hi;
    const float* ap1 = pf + (size_t)rowA1 * 36  + 2 * hi;

    v8f c00 = {}, c01 = {}, c10 = {}, c11 = {};

    // K segment 1: shape params (300)
    for (int k = 0; k < 300; k += 4) {
        v2f a0 = *(const v2f*)(as0 + k);
        v2f a1 = *(const v2f*)(as1 + k);
        v2f b0 = *(const v2f*)(sdrowA + k + 2 * hi);
        v2f b1 = *(const v2f*)(sdrowB + k + 2 * hi);
        c00 = __builtin_amdgcn_wmma_f32_16x16x4_f32(false, a0, false, b0, (short)0, c00, false, false);
        c01 = __builtin_amdgcn_wmma_f32_16x16x4_f32(false, a0, false, b1, (short)0, c01, false, false);
        c10 = __builtin_amdgcn_wmma_f32_16x16x4_f32(false, a1, false, b0, (short)0, c10, false, false);
        c11 = __builtin_amdgcn_wmma_f32_16x16x4_f32(false, a1, false, b1, (short)0, c11, false, false);
    }
    // K segment 2: expression params (100)
    for (int k = 0; k < 100; k += 4) {
        v2f a0 = *(const v2f*)(ae0 + k);
        v2f a1 = *(const v2f*)(ae1 + k);
        v2f b0 = *(const v2f*)(sdrowA + 300 + k + 2 * hi);
        v2f b1 = *(const v2f*)(sdrowB + 300 + k + 2 * hi);
        c00 = __builtin_amdgcn_wmma_f32_16x16x4_f32(false, a0, false, b0, (short)0, c00, false, false);
        c01 = __builtin_amdgcn_wmma_f32_16x16x4_f32(false, a0, false, b1, (short)0, c01, false, false);
        c10 = __builtin_amdgcn_wmma_f32_16x16x4_f32(false, a1, false, b0, (short)0, c10, false, false);
        c11 = __builtin_amdgcn_wmma_f32_16x16x4_f32(false, a1, false, b1, (short)0, c11, false, false);
    }
    // K segment 3: pose feature (36) x posedirs (K-major rows, strided B loads)
    for (int k = 0; k < 36; k += 4) {
        v2f a0 = *(const v2f*)(ap0 + k);
        v2f a1 = *(const v2f*)(ap1 + k);
        int kr = k + 2 * hi;
        v2f b0, b1;
        b0.x = pd[(size_t)kr * N3 + nca];
        b0.y = pd[(size_t)(kr + 1) * N3 + nca];
        b1.x = pd[(size_t)kr * N3 + ncb];
        b1.y = pd[(size_t)(kr + 1) * N3 + ncb];
        c00 = __builtin_amdgcn_wmma_f32_16x16x4_f32(false, a0, false, b0, (short)0, c00, false, false);
        c01 = __builtin_amdgcn_wmma_f32_16x16x4_f32(false, a0, false, b1, (short)0, c01, false, false);
        c10 = __builtin_amdgcn_wmma_f32_16x16x4_f32(false, a1, false, b0, (short)0, c10, false, false);
        c11 = __builtin_amdgcn_wmma_f32_16x16x4_f32(false, a1, false, b1, (short)0, c11, false, false);
    }

    float vta = vt[nca];
    float vtb = vt[ncb];
    if (aOK) {
        #pragma unroll
        for (int r = 0; r < 8; ++r) {
            int row = m0 + hi * 8 + r;
            out[(size_t)row * N3 + na]        = c00[r] + vta;
            out[(size_t)(row + 16) * N3 + na] = c10[r] + vta;
        }
    }
    if (bOK) {
        #pragma unroll
        for (int r = 0; r < 8; ++r) {
            int row = m0 + hi * 8 + r;
            out[(size_t)row * N3 + nb]        = c01[r] + vtb;
            out[(size_t)(row + 16) * N3 + nb] = c11[r] + vtb;
        }
    }
}

// ---------------- kernel D: per-vertex LBS (in place on vertices) + transl ----------
__global__ void flame_kD(const float* __restrict__ lw, const float* __restrict__ Aws,
                         const float* __restrict__ transl, float* __restrict__ out) {
    int idx = blockIdx.x * blockDim.x + threadIdx.x;
    if (idx >= FB * FV) return;
    int b = idx / FV;
    int v = idx - b * FV;
    const float* Ab = Aws + (size_t)b * 60;
    float T[12];
    #pragma unroll
    for (int t = 0; t < 12; ++t) T[t] = 0.f;
    for (int j = 0; j < FNJ; ++j) {
        float w = lw[(size_t)v * FNJ + j];
        const float* Aj = Ab + j * 12;
        #pragma unroll
        for (int t = 0; t < 12; ++t) T[t] += w * Aj[t];
    }
    float* pv = out + (size_t)b * N3 + (size_t)v * 3;
    float px = pv[0], py = pv[1], pz = pv[2];
    #pragma unroll
    for (int k = 0; k < 3; ++k)
        pv[k] = T[4 * k] * px + T[4 * k + 1] * py + T[4 * k + 2] * pz + T[4 * k + 3]
              + transl[b * 3 + k];
}

// ---------------- kernel E: barycentric landmarks ------------------------------------
__global__ void flame_kE(const float* __restrict__ verts, const int* __restrict__ faces,
                         const int* __restrict__ lmk_idx, const float* __restrict__ bc,
                         float* __restrict__ lm) {
    int idx = blockIdx.x * blockDim.x + threadIdx.x;
    if (idx >= FB * FNL) return;
    int b = idx / FNL;
    int l = idx - b * FNL;
    int f = lmk_idx[l];
    float a0 = 0.f, a1 = 0.f, a2 = 0.f;
    for (int t = 0; t < 3; ++t) {
        int vid = faces[f * 3 + t];
        float w = bc[l * 3 + t];
        const float* pv = verts + (size_t)b * N3 + (size_t)vid * 3;
        a0 += w * pv[0]; a1 += w * pv[1]; a2 += w * pv[2];
    }
    float* o = lm + ((size_t)b * FNL + l) * 3;
    o[0] = a0; o[1] = a1; o[2] = a2;
}

// ---------------- kernel F: flame regularizer scalar ---------------------------------
__global__ void flame_kF(const float* __restrict__ neck, const float* __restrict__ jaw,
                         const float* __restrict__ sp, const float* __restrict__ ep,
                         float* __restrict__ outreg) {
    __shared__ float red[256];
    int t = threadIdx.x;
    float acc = 0.f;
    for (int i = t; i < FB * 3;   i += 256) { float x = neck[i]; acc += 100.f  * x * x; }
    for (int i = t; i < FB * 3;   i += 256) { float x = jaw[i];  acc += 0.001f * x * x; }
    for (int i = t; i < FB * 300; i += 256) { float x = sp[i];   acc += 0.001f * x * x; }
    for (int i = t; i < FB * 100; i += 256) { float x = ep[i];   acc += 0.001f * x * x; }
    red[t] = acc;
    __syncthreads();
    for (int s = 128; s > 0; s >>= 1) {
        if (t < s) red[t] += red[t + s];
        __syncthreads();
    }
    if (t == 0) *outreg = red[0];
}

extern "C" void kernel_launch(void* const* d_in, const int* in_sizes, int n_in,
                              void* d_out, int out_size, void* d_ws, size_t ws_size,
                              hipStream_t stream) {
    const float* sp     = (const float*)d_in[0];   // shape_params    (B,300)
    const float* ep     = (const float*)d_in[1];   // expression      (B,100)
    const float* grot   = (const float*)d_in[2];   // global_rot      (B,3)
    const float* neck   = (const float*)d_in[3];   // neck_pose       (B,3)
    const float* jaw    = (const float*)d_in[4];   // jaw_pose        (B,3)
    const float* eye    = (const float*)d_in[5];   // eye_pose        (B,6)
    const float* transl = (const float*)d_in[6];   // transl          (B,3)
    const float* vt     = (const float*)d_in[7];   // v_template      (V,3)
    const float* sd     = (const float*)d_in[8];   // shapedirs       (V,3,400)
    const float* pd     = (const float*)d_in[9];   // posedirs        (36,V*3)
    const float* Jreg   = (const float*)d_in[10];  // J_regressor     (5,V)
    const float* lw     = (const float*)d_in[11];  // lbs_weights     (V,5)
    const float* bc     = (const float*)d_in[12];  // lmk_bary_coords (51,3)
    // d_in[13] = parents (compile-time constant, unused)
    const int*   faces  = (const int*)d_in[14];    // faces           (NF,3)
    const int*   lmk    = (const int*)d_in[15];    // lmk_faces_idx   (51,)

    float* out = (float*)d_out;
    float* ws  = (float*)d_ws;
    float* Js  = ws;                 // 6000 floats
    float* Jt  = ws + 6000;          // 16 (padded)
    float* pf  = ws + 6016;          // B*36 = 36864
    float* Aws = ws + 6016 + 36864;  // B*60 = 61440

    flame_kA<<<24, 256, 0, stream>>>(Jreg, vt, sd, Js, Jt);
    flame_kB<<<16, 64, 0, stream>>>(sp, ep, grot, neck, jaw, eye, Js, Jt, pf, Aws);
    flame_kC<<<(MMAC * NMAC) / 4, 128, 0, stream>>>(sp, ep, sd, pd, pf, vt, out);
    int nBV = FB * FV;
    flame_kD<<<(nBV + 255) / 256, 256, 0, stream>>>(lw, Aws, transl, out);
    int nBL = FB * FNL;
    flame_kE<<<(nBL + 255) / 256, 256, 0, stream>>>(out, faces, lmk, bc, out + (size_t)LM_OFF);
    flame_kF<<<1, 256, 0, stream>>>(neck, jaw, sp, ep, out + (size_t)REG_OFF);
}